// CoverageLoss_62457414418447
// MI455X (gfx1250) — compile-verified
//
#include <hip/hip_runtime.h>
#include <stdint.h>

#define N_LATENTS 65536
#define D 64
#define S_SAMPLES 2048
#define FARK 64
#define PUSHK 4

#define TILE 64           // latents per LDS tile
#define ROWSTRIDE 68      // floats per LDS row (64 + 4 pad -> optimal 2-phase b128 reads)
#define NTILES (N_LATENTS / TILE)   // 1024
#define WAVES_PER_BLOCK 8
#define BLOCK 256
#define SPLIT0 2          // latent-range splits, big pass  (256 blocks)
#define SPLIT1 16         // latent-range splits, far pass  (64 blocks)
#define FLT_BIG 3.402823466e38f

// ---- CDNA5 async path: global -> LDS async copies --------------------------

__device__ __forceinline__ void async_wait_all() {
  asm volatile("s_wait_asynccnt 0x0" ::: "memory");
}

__device__ __forceinline__ void async_copy_b128(uint32_t lds_addr, uint64_t gaddr) {
  // VDST = per-lane LDS byte address, VADDR = 64-bit global address (GV mode)
  asm volatile("global_load_async_to_lds_b128 %0, %1, off"
               :: "v"(lds_addr), "v"(gaddr) : "memory");
}

// ---- helpers ---------------------------------------------------------------

// keep 4 smallest of {a<=b<=c<=d, v}; stays sorted ascending
__device__ __forceinline__ void insert4(float& a, float& b, float& c, float& d, float v) {
  float m;
  m = fminf(a, v); v = fmaxf(a, v); a = m;
  m = fminf(b, v); v = fmaxf(b, v); b = m;
  m = fminf(c, v); v = fmaxf(c, v); c = m;
  d = fminf(d, v);
}

__device__ __forceinline__ float huber0(float x) {  // x >= 0 (a distance)
  return (x <= 1.0f) ? 0.5f * x * x : x - 0.5f;
}

// ---- main scan -------------------------------------------------------------
// Each wave owns TWO samples (one LDS latent read feeds both abs-diff chains).
// Block (sg, sp) scans latent range sp and emits, per sample, its 4 smallest
// distances within that range: cand[slot][SPLIT][4].
// MODE 0: slot == sample id.  MODE 1: sample id = far_idx[slot].

template <int MODE, int SPLIT, int NSAMP>
__global__ __launch_bounds__(BLOCK) void scan_kernel(
    const float* __restrict__ latents,
    const float* __restrict__ samples,
    const int* __restrict__ far_idx,
    float* __restrict__ cand) {
  __shared__ float lds[2 * TILE * ROWSTRIDE];  // 34,816 B double buffer

  const int tid  = threadIdx.x;
  const int wave = tid >> 5;
  const int lane = tid & 31;

  constexpr int NSG = NSAMP / (WAVES_PER_BLOCK * 2);
  const int sg = blockIdx.x % NSG;   // sample group
  const int sp = blockIdx.x / NSG;   // latent split
  const int slot0 = (sg * WAVES_PER_BLOCK + wave) * 2;
  const int slot1 = slot0 + 1;
  const int s0 = (MODE == 0) ? slot0 : far_idx[slot0];
  const int s1 = (MODE == 0) ? slot1 : far_idx[slot1];

  // two sample vectors resident in 128 VGPRs
  float4 sv0[16], sv1[16];
  {
    const float4* p0 = (const float4*)(samples + (size_t)s0 * D);
    const float4* p1 = (const float4*)(samples + (size_t)s1 * D);
#pragma unroll
    for (int c = 0; c < 16; ++c) { sv0[c] = p0[c]; sv1[c] = p1[c]; }
  }

  const uint32_t lds_base = (uint32_t)(uintptr_t)(&lds[0]);  // low 32 bits = LDS offset
  const uint64_t lat_base = (uint64_t)(uintptr_t)latents;

  constexpr int TPB = NTILES / SPLIT;  // tiles scanned by this block
  const int tile0 = sp * TPB;

  auto issue = [&](int tile, int buf) {
    const uint64_t gt = lat_base + (uint64_t)tile * (TILE * D * 4);
    const uint32_t lb = lds_base + (uint32_t)buf * (TILE * ROWSTRIDE * 4);
#pragma unroll
    for (int i = 0; i < (TILE * 16) / BLOCK; ++i) {  // 4 x b128 per thread per tile
      const int k = tid + i * BLOCK;
      const int r = k >> 4;
      const int c = k & 15;
      async_copy_b128(lb + (uint32_t)(r * (ROWSTRIDE * 4) + c * 16),
                      gt + (uint32_t)(r * (D * 4) + c * 16));
    }
  };

  float a0 = FLT_BIG, a1 = FLT_BIG, a2 = FLT_BIG, a3 = FLT_BIG;  // sample 0 top-4
  float b0 = FLT_BIG, b1 = FLT_BIG, b2 = FLT_BIG, b3 = FLT_BIG;  // sample 1 top-4

  issue(tile0, 0);
  async_wait_all();
  __syncthreads();

  for (int tt = 0; tt < TPB; ++tt) {
    const int cur = tt & 1;
    if (tt + 1 < TPB) issue(tile0 + tt + 1, cur ^ 1);  // prefetch next tile

    const float* buf = lds + cur * (TILE * ROWSTRIDE);
#pragma unroll
    for (int rr = 0; rr < TILE / 32; ++rr) {
      const float4* lrow = (const float4*)(buf + (lane + rr * 32) * ROWSTRIDE);
      float acc0 = 0.0f, acc1 = 0.0f;
#pragma unroll
      for (int c = 0; c < 16; ++c) {
        const float4 l = lrow[c];  // one ds_load_b128 feeds both samples
        acc0 += fabsf(sv0[c].x - l.x); acc1 += fabsf(sv1[c].x - l.x);
        acc0 += fabsf(sv0[c].y - l.y); acc1 += fabsf(sv1[c].y - l.y);
        acc0 += fabsf(sv0[c].z - l.z); acc1 += fabsf(sv1[c].z - l.z);
        acc0 += fabsf(sv0[c].w - l.w); acc1 += fabsf(sv1[c].w - l.w);
      }
      insert4(a0, a1, a2, a3, acc0);
      insert4(b0, b1, b2, b3, acc1);
    }

    async_wait_all();   // this wave's prefetch writes landed
    __syncthreads();    // everyone's did; current buffer fully consumed
  }

  // dump 32 lanes x 4 candidates per sample; lanes 0/1 merge the two samples
  float* scratch = lds;  // buffers are dead
  scratch[wave * 256 +   0 + lane * 4 + 0] = a0;
  scratch[wave * 256 +   0 + lane * 4 + 1] = a1;
  scratch[wave * 256 +   0 + lane * 4 + 2] = a2;
  scratch[wave * 256 +   0 + lane * 4 + 3] = a3;
  scratch[wave * 256 + 128 + lane * 4 + 0] = b0;
  scratch[wave * 256 + 128 + lane * 4 + 1] = b1;
  scratch[wave * 256 + 128 + lane * 4 + 2] = b2;
  scratch[wave * 256 + 128 + lane * 4 + 3] = b3;
  __syncthreads();
  if (lane < 2) {
    const float* src = scratch + wave * 256 + lane * 128;
    float c0 = FLT_BIG, c1 = FLT_BIG, c2 = FLT_BIG, c3 = FLT_BIG;
    for (int i = 0; i < 128; ++i) insert4(c0, c1, c2, c3, src[i]);
    float* dst = cand + ((size_t)(slot0 + lane) * SPLIT + sp) * 4;
    dst[0] = c0; dst[1] = c1; dst[2] = c2; dst[3] = c3;
  }
}

// ---- merge per-split candidates -> tail mean (big pass) --------------------

__global__ __launch_bounds__(BLOCK) void merge_tail_kernel(
    const float* __restrict__ cand, float* __restrict__ tail) {
  const int s = blockIdx.x * BLOCK + threadIdx.x;  // 2048 threads
  const float* src = cand + (size_t)s * SPLIT0 * 4;
  float c0 = FLT_BIG, c1 = FLT_BIG, c2 = FLT_BIG, c3 = FLT_BIG;
#pragma unroll
  for (int i = 0; i < SPLIT0 * 4; ++i) insert4(c0, c1, c2, c3, src[i]);
  tail[s] = (c0 + c1 + c2 + c3) * 0.25f;
}

// ---- top-64 selection by rank counting (matches top_k tie-break) -----------

__global__ __launch_bounds__(BLOCK) void topk_kernel(const float* __restrict__ tail,
                                                     int* __restrict__ far_idx) {
  __shared__ float v[S_SAMPLES];
  const int t = threadIdx.x;
  for (int i = 0; i < S_SAMPLES / BLOCK; ++i) v[t + i * BLOCK] = tail[t + i * BLOCK];
  __syncthreads();
  for (int i = 0; i < S_SAMPLES / BLOCK; ++i) {
    const int idx = t * (S_SAMPLES / BLOCK) + i;
    const float x = v[idx];
    int rank = 0;
    for (int j = 0; j < S_SAMPLES; ++j) {
      const float u = v[j];
      rank += (u > x) || (u == x && j < idx);
    }
    if (rank < FARK) far_idx[rank] = idx;  // ranks 0..63 distinct -> all written
  }
}

// ---- merge far candidates, Huber, deterministic final reduce ---------------

__global__ __launch_bounds__(FARK) void merge_far_kernel(
    const float* __restrict__ cand, float* __restrict__ out) {
  __shared__ float hub[FARK];
  const int t = threadIdx.x;  // 64 threads = 64 far samples
  const float* src = cand + (size_t)t * SPLIT1 * 4;
  float c0 = FLT_BIG, c1 = FLT_BIG, c2 = FLT_BIG, c3 = FLT_BIG;
  for (int i = 0; i < SPLIT1 * 4; ++i) insert4(c0, c1, c2, c3, src[i]);
  hub[t] = huber0(c0) + huber0(c1) + huber0(c2) + huber0(c3);
  __syncthreads();
  if (t == 0) {
    float s = 0.0f;
    for (int i = 0; i < FARK; ++i) s += hub[i];  // fixed order
    out[0] = s * (1.0f / (FARK * PUSHK));
  }
}

// ---- launch ----------------------------------------------------------------

extern "C" void kernel_launch(void* const* d_in, const int* in_sizes, int n_in,
                              void* d_out, int out_size, void* d_ws, size_t ws_size,
                              hipStream_t stream) {
  const float* latents = (const float*)d_in[0];   // [65536, 64] f32
  const float* samples = (const float*)d_in[1];   // [2048, 64] f32

  float* wsf     = (float*)d_ws;
  float* cand0   = wsf;                                   // 2048*2*4 = 16384 floats
  float* tail    = cand0 + (size_t)S_SAMPLES * SPLIT0 * 4;  // 2048
  int*   far_idx = (int*)(tail + S_SAMPLES);                // 64
  float* cand1   = (float*)(far_idx + FARK);                // 64*16*4 = 4096
  float* out     = (float*)d_out;

  // phase 1: 128 sample-groups x 2 latent-halves = 256 blocks
  scan_kernel<0, SPLIT0, S_SAMPLES>
      <<<(S_SAMPLES / (WAVES_PER_BLOCK * 2)) * SPLIT0, BLOCK, 0, stream>>>(
          latents, samples, (const int*)nullptr, cand0);
  merge_tail_kernel<<<S_SAMPLES / BLOCK, BLOCK, 0, stream>>>(cand0, tail);
  topk_kernel<<<1, BLOCK, 0, stream>>>(tail, far_idx);

  // phase 2: 4 sample-groups x 16 latent-splits = 64 blocks
  scan_kernel<1, SPLIT1, FARK>
      <<<(FARK / (WAVES_PER_BLOCK * 2)) * SPLIT1, BLOCK, 0, stream>>>(
          latents, samples, far_idx, cand1);
  merge_far_kernel<<<1, FARK, 0, stream>>>(cand1, out);
}